// DegreePrediction_11527692223141
// MI455X (gfx1250) — compile-verified
//
#include <hip/hip_runtime.h>
#include <hip/hip_bf16.h>

#define NMAT 4096   // 64*64 matrices
#define N    64

typedef __attribute__((ext_vector_type(2))) float v2f;
typedef __attribute__((ext_vector_type(8))) float v8f;

// ---------------------------------------------------------------------------
// Kernel 1: one workgroup (64 threads = 2 waves) per 64x64 matrix.
//   A = weights_r*r_zeros + r_const - 1*I   (built directly in LDS, coalesced)
//   P*A = L*U  (partial pivoting, thread-per-row trailing update)
//   inverse iteration (4 sweeps) -> eigenvector for eigenvalue closest to 1
//   P_out[m][n] = T[m] * v[n]/v[s],  T = x*weights_t*r_const[s,t,s,s]
// ---------------------------------------------------------------------------
__global__ __launch_bounds__(64) void eig_invit_kernel(
    const float* __restrict__ x,
    const float* __restrict__ r_zeros,
    const float* __restrict__ r_const,
    const float* __restrict__ weights_t,
    const float* __restrict__ weights_r,
    float* __restrict__ Pout)
{
    __shared__ float A[N][N + 1];   // +1 pad: stride 65 -> bank-conflict-free column walks
    __shared__ float b[N];
    __shared__ int   piv[N];
    __shared__ float scal;

    const int tid  = threadIdx.x;
    const int m    = blockIdx.x;       // m = s*64 + t
    const int s    = m >> 6;
    const size_t base = (size_t)m * (N * N);

    // Build shifted matrix in LDS; loop over rows, lanes over columns (coalesced).
    for (int r = 0; r < N; ++r) {
        size_t idx = base + (size_t)r * N + tid;
        float a = weights_r[idx] * r_zeros[idx] + r_const[idx];
        if (r == tid) a -= 1.0f;       // shift by target eigenvalue 1
        A[r][tid] = a;
    }
    b[tid] = 1.0f;                     // inverse-iteration start vector
    __syncthreads();

    // LU factorization with partial pivoting.
    for (int k = 0; k < N; ++k) {
        if (tid == 0) {
            int p = k; float best = fabsf(A[k][k]);
            for (int i = k + 1; i < N; ++i) {
                float v = fabsf(A[i][k]);
                if (v > best) { best = v; p = i; }
            }
            piv[k] = p;
        }
        __syncthreads();
        const int p = piv[k];
        if (p != k) {                  // swap rows k<->p, lanes over columns
            float t0 = A[k][tid];
            A[k][tid] = A[p][tid];
            A[p][tid] = t0;
        }
        __syncthreads();
        float pv = A[k][k];
        if (fabsf(pv) < 1e-30f) pv = (pv < 0.0f) ? -1e-30f : 1e-30f;
        if (tid > k) {                 // thread-per-row rank-1 trailing update
            float mlt = A[tid][k] / pv;
            A[tid][k] = mlt;
            for (int j = k + 1; j < N; ++j)
                A[tid][j] -= mlt * A[k][j];
        }
        __syncthreads();
    }

    // Inverse iteration: v <- normalize((A - I)^{-1} v)
    for (int it = 0; it < 4; ++it) {
        if (tid == 0) {                // apply recorded row swaps, in order
            for (int k = 0; k < N; ++k) {
                int p = piv[k];
                if (p != k) { float t0 = b[k]; b[k] = b[p]; b[p] = t0; }
            }
        }
        __syncthreads();
        // Forward solve (unit-lower L), column sweep.
        for (int k = 0; k < N - 1; ++k) {
            if (tid > k) b[tid] -= A[tid][k] * b[k];
            __syncthreads();
        }
        // Back solve (U), column sweep.
        for (int k = N - 1; k >= 0; --k) {
            if (tid == k) {
                float d = A[k][k];
                if (fabsf(d) < 1e-30f) d = (d < 0.0f) ? -1e-30f : 1e-30f;
                b[k] = b[k] / d;
            }
            __syncthreads();
            if (tid < k) b[tid] -= A[tid][k] * b[k];
            __syncthreads();
        }
        // Normalize by inf-norm to avoid overflow.
        if (tid == 0) {
            float mx = 0.0f;
            for (int i = 0; i < N; ++i) mx = fmaxf(mx, fabsf(b[i]));
            scal = (mx > 0.0f) ? (1.0f / mx) : 1.0f;
        }
        __syncthreads();
        b[tid] *= scal;
        __syncthreads();
    }

    // vn = v / v[s]; T = x*weights_t*rc_diag; partial row = T * vn.
    float denom = b[s];
    if (fabsf(denom) < 1e-30f) denom = (denom < 0.0f) ? -1e-30f : 1e-30f;
    const float rc_diag = r_const[base + (size_t)s * N + s];
    const float T = x[m] * weights_t[m] * rc_diag;
    Pout[(size_t)m * N + tid] = T * (b[tid] / denom);
}

// ---------------------------------------------------------------------------
// Kernel 2: out[j] = sum_m Pout[m][j] via V_WMMA_F32_16X16X4_F32.
// A-fragment = all ones  =>  D[i][j] = sum_k B[k][j]  (column sums).
// With A == 1 the result does not depend on how the hardware assigns K slots
// inside the B fragment; only the lane->N mapping of C/D matters (documented:
// lanes 0-15 -> N = lane). 4 waves, each owning a 16-column slice, chain 1024
// accumulating WMMAs over the 4096 rows.
// ---------------------------------------------------------------------------
__global__ __launch_bounds__(128) void reduce_wmma_kernel(
    const float* __restrict__ Pout, float* __restrict__ out)
{
    const int lane    = threadIdx.x & 31;
    const int wave    = threadIdx.x >> 5;        // 0..3
    const int colbase = wave * 16;
    const int col     = colbase + (lane & 15);
    const int rowoff  = lane >> 4;               // 0 for lanes 0-15, 1 for 16-31

    v2f a; a.x = 1.0f; a.y = 1.0f;               // ones A-fragment (16x4)
    v8f c = {};                                  // zero accumulator

    for (int m0 = 0; m0 < NMAT; m0 += 4) {
        v2f bf;
        bf.x = Pout[(size_t)(m0 + rowoff)     * N + col];
        bf.y = Pout[(size_t)(m0 + 2 + rowoff) * N + col];
        // 8 args: (neg_a, A, neg_b, B, c_mod, C, reuse_a, reuse_b)
        c = __builtin_amdgcn_wmma_f32_16x16x4_f32(
                false, a, false, bf, (short)0, c, false, false);
    }

    // Every row of D equals the column sums; take VGPR0, lanes 0-15 (N = lane).
    if (lane < 16) out[col] = c[0];
}

extern "C" void kernel_launch(void* const* d_in, const int* in_sizes, int n_in,
                              void* d_out, int out_size, void* d_ws, size_t ws_size,
                              hipStream_t stream) {
    (void)in_sizes; (void)n_in; (void)out_size; (void)ws_size;
    const float* x         = (const float*)d_in[0];
    const float* r_zeros   = (const float*)d_in[1];
    const float* r_const   = (const float*)d_in[2];
    const float* weights_t = (const float*)d_in[3];
    const float* weights_r = (const float*)d_in[4];
    float* out = (float*)d_out;
    float* P   = (float*)d_ws;   // 4096 * 64 floats = 1 MiB scratch

    eig_invit_kernel<<<NMAT, 64, 0, stream>>>(x, r_zeros, r_const, weights_t, weights_r, P);
    reduce_wmma_kernel<<<1, 128, 0, stream>>>(P, out);
}